// QConvNet_60129542144172
// MI455X (gfx1250) — compile-verified
//
#include <hip/hip_runtime.h>
#include <math.h>

// ---------------------------------------------------------------------------
// CDNA5 (gfx1250) fp32 WMMA fragments: V_WMMA_F32_16X16X4_F32
//   A (16x4 f32):  2 VGPRs. lane%16 = M row; lanes 0-15 hold K=0,1; lanes 16-31 hold K=2,3
//   B (4x16 f32):  2 VGPRs. N = lane%16; VGPR0 holds K=0 (low half) / K=2 (high half), VGPR1 K=1/3
//   C/D (16x16):   8 VGPRs. VGPR v: M = v + 8*(lane>=16), N = lane%16
// ---------------------------------------------------------------------------
typedef __attribute__((ext_vector_type(2))) float v2f;
typedef __attribute__((ext_vector_type(8))) float v8f;

// Hamilton-product block tables (identical numeric table for conv rows and linear cols)
__device__ const int   qsel[4][4] = {{0,1,2,3},{1,0,3,2},{2,3,0,1},{3,2,1,0}};
__device__ const float qsgn[4][4] = {{1.f,-1.f,-1.f,-1.f},
                                     {1.f, 1.f,-1.f, 1.f},
                                     {1.f, 1.f, 1.f,-1.f},
                                     {1.f,-1.f, 1.f, 1.f}};

// ---------------------------------------------------------------------------
// Weight expansion: comps (OG, IG, 3, 3) -> Wt[K = IG*4*9][COUT = OG*4]
// W[o, ci, r, s] with o = ob*OG + oo, ci = ib*IG + ii:
//   W = qsgn[ob][ib] * comp[qsel[ob][ib]][oo, ii, r, s]
// Stored transposed (K-major) so conv B-fragment loads are coalesced.
// ---------------------------------------------------------------------------
template<int OG, int IG>
__global__ void expand_qconv(const float* __restrict__ cr, const float* __restrict__ ci_,
                             const float* __restrict__ cj, const float* __restrict__ ck,
                             float* __restrict__ Wt) {
    const int COUT = OG * 4;
    const int CIN  = IG * 4;
    int idx = blockIdx.x * blockDim.x + threadIdx.x;
    int total = COUT * CIN * 9;
    if (idx >= total) return;
    int o   = idx / (CIN * 9);
    int rem = idx % (CIN * 9);
    int ci  = rem / 9;
    int rs  = rem % 9;
    int ob = o / OG, oo = o % OG;
    int ib = ci / IG, ii = ci % IG;
    const float* comps[4] = {cr, ci_, cj, ck};
    float v = qsgn[ob][ib] * comps[qsel[ob][ib]][(oo * IG + ii) * 9 + rs];
    Wt[(ci * 9 + rs) * COUT + o] = v;   // K = ci*9+rs
}

// comps (512, 4) = (in/4, out/4) -> Wl[2048][16]
__global__ void expand_qlin(const float* __restrict__ cr, const float* __restrict__ ci_,
                            const float* __restrict__ cj, const float* __restrict__ ck,
                            float* __restrict__ Wl) {
    int idx = blockIdx.x * blockDim.x + threadIdx.x;
    if (idx >= 2048 * 16) return;
    int kin = idx / 16;          // input index
    int o   = idx % 16;          // output index
    int ib = kin / 512, ii = kin % 512;
    int ob = o / 4,     oo = o % 4;
    const float* comps[4] = {cr, ci_, cj, ck};
    Wl[kin * 16 + o] = qsgn[ob][ib] * comps[qsel[ob][ib]][ii * 4 + oo];
}

__global__ void zero_kernel(float* p, int n) {
    int t = blockIdx.x * 256 + threadIdx.x;
    if (t < n) p[t] = 0.0f;
}

// ---------------------------------------------------------------------------
// Global average pool over H*W=1024 per (b,c).  The 210 MB bandwidth pass.
// ---------------------------------------------------------------------------
__global__ void gap_kernel(const float* __restrict__ x, float* __restrict__ pooled) {
    int bc = blockIdx.x;
    const float4* p = (const float4*)(x + (size_t)bc * 1024);
    float4 v = p[threadIdx.x];
    float s = v.x + v.y + v.z + v.w;
    __shared__ float sm[256];
    sm[threadIdx.x] = s;
    __syncthreads();
    for (int st = 128; st > 0; st >>= 1) {
        if (threadIdx.x < st) sm[threadIdx.x] += sm[threadIdx.x + st];
        __syncthreads();
    }
    if (threadIdx.x == 0) pooled[bc] = sm[0] * (1.0f / 1024.0f);
}

// top-3 channels of softmax(w*pooled) == top-3 of w*pooled (softmax monotone).
// Tie-break: lower index wins (matches lax.top_k / torch.topk).
__global__ void top3_kernel(const float* __restrict__ pooled, const float* __restrict__ w_eca,
                            int* __restrict__ idx3, int C) {
    int b = blockIdx.x;
    int t = threadIdx.x;
    __shared__ float sv[256];
    __shared__ int   si[256];
    float w = w_eca[0];
    float v = (t < C) ? w * pooled[b * C + t] : -INFINITY;
    for (int pick = 0; pick < 3; pick++) {
        sv[t] = v; si[t] = t;
        __syncthreads();
        for (int st = 128; st > 0; st >>= 1) {
            if (t < st) {
                float v2 = sv[t + st]; int i2 = si[t + st];
                if (v2 > sv[t] || (v2 == sv[t] && i2 < si[t])) { sv[t] = v2; si[t] = i2; }
            }
            __syncthreads();
        }
        if (t == 0) idx3[b * 3 + pick] = si[0];
        if (t == si[0]) v = -INFINITY;
        __syncthreads();
    }
}

// q[b,0,:,:] = 0 ; q[b,1+t,:,:] = x[b, idx[b,t], :, :]
__global__ void gather_q(const float* __restrict__ x, const int* __restrict__ idx3,
                         float* __restrict__ q, int total) {
    int tid = blockIdx.x * 256 + threadIdx.x;
    if (tid >= total) return;
    int p = tid & 1023;
    int c = (tid >> 10) & 3;
    int b = tid >> 12;
    q[tid] = (c == 0) ? 0.0f
                      : x[((size_t)b * 200 + idx3[b * 3 + (c - 1)]) * 1024 + p];
}

// ---------------------------------------------------------------------------
// 3x3 same-pad conv as implicit GEMM with fp32 WMMA.
//   M = B*H*W pixels (16 per block, im2row staged in LDS, shared by all waves)
//   N = COUT (one 16-wide tile per wave), K = CIN*9 (always a multiple of 4)
// ---------------------------------------------------------------------------
template<int CIN, int COUT, int H, int W>
__global__ __launch_bounds__(32 * (COUT / 16))
void qconv3x3_wmma(const float* __restrict__ in, const float* __restrict__ Wt,
                   const float* __restrict__ bias, float* __restrict__ out) {
    constexpr int K  = CIN * 9;
    constexpr int NT = COUT / 16;
    constexpr int HW = H * W;
    constexpr int LK = K + 1;                 // odd stride -> conflict-free LDS rows
    __shared__ float As[16 * LK];

    const int m0 = blockIdx.x * 16;           // HW % 16 == 0, so one batch per tile
    const int b  = m0 / HW;
    const int p0 = m0 % HW;
    const int tid = threadIdx.y * 32 + threadIdx.x;

    // im2row stage: As[row][k] = in[b, ci, y+dy, x+dx] (0 outside)
    for (int e = tid; e < 16 * K; e += 32 * NT) {
        int mrow = e / K, k = e % K;
        int ci = k / 9, rs = k % 9;
        int p = p0 + mrow;
        int y = p / W + rs / 3 - 1;
        int x = p % W + rs % 3 - 1;
        float v = 0.0f;
        if ((unsigned)y < (unsigned)H && (unsigned)x < (unsigned)W)
            v = in[(((size_t)b * CIN + ci) * H + y) * W + x];
        As[mrow * LK + k] = v;
    }
    __syncthreads();

    const int lane = threadIdx.x;
    const int half = lane >> 4;
    const int mr   = lane & 15;
    const int n0   = threadIdx.y * 16;
    const float* wcol = Wt + n0 + mr;         // B fragment column for this lane

    v8f acc = {};
    for (int k0 = 0; k0 < K; k0 += 4) {
        v2f a, bf;
        a.x = As[mr * LK + k0 + 2 * half];
        a.y = As[mr * LK + k0 + 2 * half + 1];
        bf.x = wcol[(size_t)(k0 + 2 * half) * COUT];
        bf.y = wcol[(size_t)(k0 + 2 * half + 1) * COUT];
        if (k0 + 4 < K) __builtin_prefetch(wcol + (size_t)(k0 + 4) * COUT, 0, 0);
        acc = __builtin_amdgcn_wmma_f32_16x16x4_f32(false, a, false, bf,
                                                    (short)0, acc, false, false);
    }

#pragma unroll
    for (int v = 0; v < 8; v++) {
        int Mi = v + half * 8;                // D row
        int c  = n0 + mr;                     // D col = output channel
        out[((size_t)b * COUT + c) * HW + p0 + Mi] = acc[v] + bias[c];
    }
}

// ---------------------------------------------------------------------------
// Quaternion BN stats: per-component Sum(x), Sum(x^2) over the WHOLE tensor.
// One block per (b, comp) contiguous slab; float atomics into stats[8].
// ---------------------------------------------------------------------------
__global__ void stats_kernel(const float* __restrict__ h, float* __restrict__ stats,
                             int C4HW) {
    int slab = blockIdx.x;                    // b*4 + comp
    int comp = slab & 3;
    const float* base = h + (size_t)slab * C4HW;
    float s = 0.f, s2 = 0.f;
    for (int e = threadIdx.x; e < C4HW; e += 256) {
        float v = base[e];
        s += v; s2 += v * v;
    }
    __shared__ float sm[256], sq[256];
    sm[threadIdx.x] = s; sq[threadIdx.x] = s2;
    __syncthreads();
    for (int st = 128; st > 0; st >>= 1) {
        if (threadIdx.x < st) {
            sm[threadIdx.x] += sm[threadIdx.x + st];
            sq[threadIdx.x] += sq[threadIdx.x + st];
        }
        __syncthreads();
    }
    if (threadIdx.x == 0) {
        atomicAdd(&stats[comp * 2 + 0], sm[0]);
        atomicAdd(&stats[comp * 2 + 1], sq[0]);
    }
}

// Fused quaternion-BN + ReLU + 2x2 avg pool.
// var = sum_c (E[c^2] - mean_c^2); out = relu(gamma[c%C4]*(x-mean)*rsqrt(var+eps)+beta[c])
template<int C, int H, int W>
__global__ void bnrelupool(const float* __restrict__ h, const float* __restrict__ stats,
                           const float* __restrict__ gamma, const float* __restrict__ beta,
                           float* __restrict__ out, int total, int B) {
    constexpr int C4 = C / 4;
    constexpr int Ho = H / 2, Wo = W / 2;
    int tid = blockIdx.x * 256 + threadIdx.x;
    if (tid >= total) return;
    int xo = tid % Wo;
    int yo = (tid / Wo) % Ho;
    int c  = (tid / (Wo * Ho)) % C;
    int b  = tid / (Wo * Ho * C);

    float Ninv = 1.0f / ((float)B * (float)C4 * (float)(H * W));
    float m0 = stats[0] * Ninv, m1 = stats[2] * Ninv;
    float m2 = stats[4] * Ninv, m3 = stats[6] * Ninv;
    float var = (stats[1] * Ninv - m0 * m0) + (stats[3] * Ninv - m1 * m1) +
                (stats[5] * Ninv - m2 * m2) + (stats[7] * Ninv - m3 * m3);
    float inv = rsqrtf(var + 1e-5f);

    int comp = c / C4, sub = c % C4;
    float mean = (comp == 0) ? m0 : (comp == 1) ? m1 : (comp == 2) ? m2 : m3;
    float g  = gamma[sub] * inv;
    float bt = beta[c];

    const float* src = h + ((size_t)(b * C + c) * H + yo * 2) * W + xo * 2;
    float acc = fmaxf(0.f, g * (src[0]     - mean) + bt)
              + fmaxf(0.f, g * (src[1]     - mean) + bt)
              + fmaxf(0.f, g * (src[W]     - mean) + bt)
              + fmaxf(0.f, g * (src[W + 1] - mean) + bt);
    out[tid] = acc * 0.25f;
}

// ---------------------------------------------------------------------------
// FC: [B,2048] @ [2048,16] + bias, fp32 WMMA, one wave per 16-row tile.
// ---------------------------------------------------------------------------
__global__ __launch_bounds__(32)
void fc_wmma(const float* __restrict__ A, const float* __restrict__ Wl,
             const float* __restrict__ bias, float* __restrict__ out, int Kdim) {
    const int m0   = blockIdx.x * 16;
    const int lane = threadIdx.x;
    const int half = lane >> 4;
    const int mr   = lane & 15;
    const float* arow = A + (size_t)(m0 + mr) * Kdim;
    v8f acc = {};
    for (int k0 = 0; k0 < Kdim; k0 += 4) {
        v2f a, bf;
        a.x = arow[k0 + 2 * half];
        a.y = arow[k0 + 2 * half + 1];
        bf.x = Wl[(k0 + 2 * half) * 16 + mr];
        bf.y = Wl[(k0 + 2 * half + 1) * 16 + mr];
        acc = __builtin_amdgcn_wmma_f32_16x16x4_f32(false, a, false, bf,
                                                    (short)0, acc, false, false);
    }
#pragma unroll
    for (int v = 0; v < 8; v++) {
        int Mi = v + half * 8;
        out[(size_t)(m0 + Mi) * 16 + mr] = acc[v] + bias[mr];
    }
}

// ---------------------------------------------------------------------------
extern "C" void kernel_launch(void* const* d_in, const int* in_sizes, int n_in,
                              void* d_out, int out_size, void* d_ws, size_t ws_size,
                              hipStream_t stream) {
    const float* x     = (const float*)d_in[0];
    const float* w_eca = (const float*)d_in[1];
    const float* l1r = (const float*)d_in[2],  *l1i = (const float*)d_in[3];
    const float* l1j = (const float*)d_in[4],  *l1k = (const float*)d_in[5];
    const float* l1b = (const float*)d_in[6];
    const float* g1  = (const float*)d_in[7],  *b1  = (const float*)d_in[8];
    const float* l2r = (const float*)d_in[9],  *l2i = (const float*)d_in[10];
    const float* l2j = (const float*)d_in[11], *l2k = (const float*)d_in[12];
    const float* l2b = (const float*)d_in[13];
    const float* g2  = (const float*)d_in[14], *b2  = (const float*)d_in[15];
    const float* l3r = (const float*)d_in[16], *l3i = (const float*)d_in[17];
    const float* l3j = (const float*)d_in[18], *l3k = (const float*)d_in[19];
    const float* l3b = (const float*)d_in[20];
    const float* g3  = (const float*)d_in[21], *b3  = (const float*)d_in[22];
    const float* fcr = (const float*)d_in[23], *fci = (const float*)d_in[24];
    const float* fcj = (const float*)d_in[25], *fck = (const float*)d_in[26];
    const float* fcb = (const float*)d_in[27];

    const int B = in_sizes[0] / (200 * 32 * 32);

    float* ws = (float*)d_ws;
    size_t off = 0;
    auto take = [&](size_t n) -> float* {
        float* p = ws + off;
        off += (n + 15) & ~(size_t)15;
        return p;
    };
    float* pooled = take((size_t)B * 200);
    int*   idx3   = (int*)take((size_t)B * 4);
    float* q      = take((size_t)B * 4  * 1024);
    float* h1     = take((size_t)B * 16 * 1024);
    float* h1p    = take((size_t)B * 16 * 256);
    float* h2     = take((size_t)B * 64 * 256);
    float* h2p    = take((size_t)B * 64 * 64);
    float* h3     = take((size_t)B * 128 * 64);
    float* h3p    = take((size_t)B * 128 * 16);
    float* Wt1    = take(36 * 16);
    float* Wt2    = take(144 * 64);
    float* Wt3    = take(576 * 128);
    float* Wfc    = take(2048 * 16);
    float* stats  = take(32);          // 3 layers x (sum,sumsq) x 4 comps

    // --- prep: weight expansion + zero stats (deterministic every call)
    expand_qconv<4, 1>  <<<dim3((576    + 255) / 256), dim3(256), 0, stream>>>(l1r, l1i, l1j, l1k, Wt1);
    expand_qconv<16, 4> <<<dim3((9216   + 255) / 256), dim3(256), 0, stream>>>(l2r, l2i, l2j, l2k, Wt2);
    expand_qconv<32, 16><<<dim3((73728  + 255) / 256), dim3(256), 0, stream>>>(l3r, l3i, l3j, l3k, Wt3);
    expand_qlin         <<<dim3((32768  + 255) / 256), dim3(256), 0, stream>>>(fcr, fci, fcj, fck, Wfc);
    zero_kernel<<<dim3(1), dim3(32), 0, stream>>>(stats, 24);

    // --- pre_press: GAP (the 210MB pass) -> top3 -> gather
    gap_kernel <<<dim3(B * 200), dim3(256), 0, stream>>>(x, pooled);
    top3_kernel<<<dim3(B),       dim3(256), 0, stream>>>(pooled, w_eca, idx3, 200);
    gather_q   <<<dim3((B * 4096 + 255) / 256), dim3(256), 0, stream>>>(x, idx3, q, B * 4096);

    // --- layer1: conv(4->16, 32x32) + qbn + relu + pool
    qconv3x3_wmma<4, 16, 32, 32><<<dim3(B * 64), dim3(32, 1), 0, stream>>>(q, Wt1, l1b, h1);
    stats_kernel<<<dim3(B * 4), dim3(256), 0, stream>>>(h1, stats + 0, 4 * 1024);
    bnrelupool<16, 32, 32><<<dim3(B * 16), dim3(256), 0, stream>>>(h1, stats + 0, g1, b1, h1p, B * 4096, B);

    // --- layer2: conv(16->64, 16x16) + qbn + relu + pool
    qconv3x3_wmma<16, 64, 16, 16><<<dim3(B * 16), dim3(32, 4), 0, stream>>>(h1p, Wt2, l2b, h2);
    stats_kernel<<<dim3(B * 4), dim3(256), 0, stream>>>(h2, stats + 8, 16 * 256);
    bnrelupool<64, 16, 16><<<dim3(B * 16), dim3(256), 0, stream>>>(h2, stats + 8, g2, b2, h2p, B * 4096, B);

    // --- layer3: conv(64->128, 8x8) + qbn + relu + pool
    qconv3x3_wmma<64, 128, 8, 8><<<dim3(B * 4), dim3(32, 8), 0, stream>>>(h2p, Wt3, l3b, h3);
    stats_kernel<<<dim3(B * 4), dim3(256), 0, stream>>>(h3, stats + 16, 32 * 64);
    bnrelupool<128, 8, 8><<<dim3(B * 8), dim3(256), 0, stream>>>(h3, stats + 16, g3, b3, h3p, B * 2048, B);

    // --- fc: [B,2048] @ [2048,16] + bias
    fc_wmma<<<dim3(B / 16), dim3(32), 0, stream>>>(h3p, Wfc, fcb, (float*)d_out, 2048);
}